// Downsample_8409545966243
// MI455X (gfx1250) — compile-verified
//
#include <hip/hip_runtime.h>
#include <cstdint>

#ifndef __has_builtin
#define __has_builtin(x) 0
#endif

#if __has_builtin(__builtin_amdgcn_global_load_async_to_lds_b128)
#define HAVE_ASYNC_G2L 1
#else
#define HAVE_ASYNC_G2L 0
#endif

typedef float v4f __attribute__((ext_vector_type(4)));
typedef int   v4i __attribute__((ext_vector_type(4)));

#if HAVE_ASYNC_G2L
typedef __attribute__((address_space(1))) v4i* as1_v4i;
typedef __attribute__((address_space(3))) v4i* as3_v4i;
#endif

namespace {
constexpr int H_IN = 256, W_IN = 256, H_OUT = 128, W_OUT = 128;
constexpr int OUT_ROWS  = 16;               // output rows per block
constexpr int TILE_ROWS = 2 * OUT_ROWS + 2; // 34 input rows staged
constexpr int FRONT = 4;                    // front pad (col0 stays 16B-aligned; col -1 lives at FRONT-1)
constexpr int LDS_STRIDE = FRONT + W_IN + 4; // 264 floats per row
}

// Stage one 16-byte chunk: HBM -> LDS directly (CDNA5 async path), or fallback.
__device__ __forceinline__ void stage_chunk_b128(const float* g, float* lds_dst) {
#if HAVE_ASYNC_G2L
  __builtin_amdgcn_global_load_async_to_lds_b128(
      (as1_v4i)(uintptr_t)g,
      (as3_v4i)(uint32_t)(uintptr_t)lds_dst,   // low 32 bits of generic LDS addr = LDS offset
      0, 0);
#else
  const v4f v = *reinterpret_cast<const v4f*>(g);
  *reinterpret_cast<v4f*>(lds_dst) = v;
#endif
}

__device__ __forceinline__ void wait_async_done() {
#if HAVE_ASYNC_G2L
#if __has_builtin(__builtin_amdgcn_s_wait_asynccnt)
  __builtin_amdgcn_s_wait_asynccnt(0);
#else
  asm volatile("s_wait_asynccnt 0" ::: "memory");
#endif
#endif
}

__global__ __launch_bounds__(256)
void downsample2x_fir4(const float* __restrict__ in,
                       const float* __restrict__ kern,
                       float* __restrict__ out) {
  __shared__ float tile[TILE_ROWS * LDS_STRIDE];

  const int tid   = threadIdx.x;
  const unsigned bid = blockIdx.x;
  const int plane = bid >> 3;   // (b*C + c)
  const int slab  = bid & 7;
  const int r0 = slab * OUT_ROWS;      // first output row of slab
  const int g0 = 2 * r0 - 1;           // first input row needed (may be -1)
  const float* src = in + (size_t)plane * ((size_t)H_IN * W_IN);

  // Zero the two pad columns (col -1 and col 256) of every staged row.
  if (tid < TILE_ROWS) {
    tile[tid * LDS_STRIDE + (FRONT - 1)]  = 0.f;
    tile[tid * LDS_STRIDE + FRONT + W_IN] = 0.f;
  }

  // Stage 34 rows x 256 f32 as 64 b128 chunks/row. 34*64 = 2176 chunks.
  #pragma unroll
  for (int it = 0; it < 9; ++it) {
    const int idx = tid + it * 256;
    if (idx < TILE_ROWS * 64) {
      const int row = idx >> 6;
      const int c4  = (idx & 63) * 4;
      float* dst = &tile[row * LDS_STRIDE + FRONT + c4];
      const int g = g0 + row;
      if ((unsigned)g < (unsigned)H_IN) {
        stage_chunk_b128(src + (size_t)g * W_IN + c4, dst);
      } else {
        *reinterpret_cast<v4f*>(dst) = (v4f){0.f, 0.f, 0.f, 0.f}; // H zero-pad rows
      }
    }
  }

  wait_async_done();   // my wave's async HBM->LDS writes landed
  __syncthreads();     // everyone's landed

  // True convolution => flipped kernel taps (uniform -> scalar loads).
  float t[16];
  #pragma unroll
  for (int a = 0; a < 4; ++a)
    #pragma unroll
    for (int b = 0; b < 4; ++b)
      t[a * 4 + b] = kern[(3 - a) * 4 + (3 - b)];

  // Thread -> (output row i in tile, 8-wide column group k). wo = 8k..8k+7.
  const int i = tid >> 4;
  const int k = tid & 15;

  float acc[8] = {0.f, 0.f, 0.f, 0.f, 0.f, 0.f, 0.f, 0.f};

  // Output row r0+i needs local input rows 2i..2i+3; cols 16k-1 .. 16k+16.
  #pragma unroll
  for (int a = 0; a < 4; ++a) {
    const float* rp = &tile[(2 * i + a) * LDS_STRIDE + 16 * k]; // float idx 16k == col 16k-4
    float f[24];
    #pragma unroll
    for (int q = 0; q < 6; ++q) {   // 6x ds_load_b128, 16B aligned, bank-balanced
      const v4f v = *reinterpret_cast<const v4f*>(rp + 4 * q);
      f[4 * q + 0] = v.x; f[4 * q + 1] = v.y; f[4 * q + 2] = v.z; f[4 * q + 3] = v.w;
    }
    #pragma unroll
    for (int d = 0; d < 8; ++d)
      #pragma unroll
      for (int b = 0; b < 4; ++b)
        acc[d] = fmaf(t[a * 4 + b], f[2 * d + b + 3], acc[d]); // col 2(8k+d)-1+b -> f[2d+b+3]
  }

  // Coalesced nontemporal stores: 2x b128 per thread, lanes contiguous.
  float* dstp = out + ((size_t)plane * H_OUT + (r0 + i)) * (size_t)W_OUT + 8 * k;
  const v4f o0 = {acc[0], acc[1], acc[2], acc[3]};
  const v4f o1 = {acc[4], acc[5], acc[6], acc[7]};
  __builtin_nontemporal_store(o0, reinterpret_cast<v4f*>(dstp));
  __builtin_nontemporal_store(o1, reinterpret_cast<v4f*>(dstp) + 1);
}

extern "C" void kernel_launch(void* const* d_in, const int* in_sizes, int n_in,
                              void* d_out, int out_size, void* d_ws, size_t ws_size,
                              hipStream_t stream) {
  const float* in   = (const float*)d_in[0];
  const float* kern = (const float*)d_in[1];
  float* out        = (float*)d_out;

  const int planes = in_sizes[0] / (H_IN * W_IN);          // 8*256 = 2048
  const int blocks = planes * (H_OUT / OUT_ROWS);          // 2048 * 8 = 16384
  downsample2x_fir4<<<dim3(blocks), dim3(256), 0, stream>>>(in, kern, out);
}